// GCN_82463372083253
// MI455X (gfx1250) — compile-verified
//
#include <hip/hip_runtime.h>
#include <hip/hip_bf16.h>

typedef __attribute__((ext_vector_type(2))) float v2f;
typedef __attribute__((ext_vector_type(8))) float v8f;

#define FEAT 128

// ---------------- degree kernels ----------------
__global__ void deg_init(float* __restrict__ dis, int n) {
    int i = blockIdx.x * blockDim.x + threadIdx.x;
    if (i < n) dis[i] = 1.0f;   // self-loop contributes 1 to in-degree
}

__global__ void deg_accum(const long long* __restrict__ dst, float* __restrict__ dis, int e) {
    int i = blockIdx.x * blockDim.x + threadIdx.x;
    if (i < e) atomicAdd(&dis[(int)dst[i]], 1.0f);
}

__global__ void deg_finalize(float* __restrict__ dis, int n) {
    int i = blockIdx.x * blockDim.x + threadIdx.x;
    if (i < n) dis[i] = rsqrtf(dis[i]);   // deg >= 1 always (self-loops)
}

// ---------------- f32 WMMA GEMM: H = X @ W^T ----------------
// X: [n,128] row-major, W: [128(out),128(in)] row-major, H: [n,128]
// Block: 256 threads = 8 wave32. Block handles 16 rows; wave w handles out cols [16w,16w+16).
__global__ void __launch_bounds__(256) gemm_wmma_f32(const float* __restrict__ X,
                                                     const float* __restrict__ W,
                                                     float* __restrict__ H, int n) {
    const int wave = threadIdx.x >> 5;
    const int lane = threadIdx.x & 31;
    const int half = lane >> 4;       // 0: K pair {0,1}, 1: K pair {2,3}
    const int l15  = lane & 15;
    const int row0 = blockIdx.x * 16;

    int arow = row0 + l15;
    if (arow >= n) arow = n - 1;                       // clamp loads; EXEC stays all-1s
    const float* xrow = X + (size_t)arow * FEAT;       // A: M = l15
    const float* wrow = W + (size_t)(wave * 16 + l15) * FEAT;  // B: N = l15 -> W row

    v8f c = {};
    #pragma unroll
    for (int k = 0; k < FEAT; k += 4) {
        const int ka = k + half * 2;
        v2f a = *(const v2f*)(xrow + ka);   // A[m][ka], A[m][ka+1]
        v2f b = *(const v2f*)(wrow + ka);   // B[ka][nn] = W[n_out][ka], and ka+1
        c = __builtin_amdgcn_wmma_f32_16x16x4_f32(false, a, false, b,
                                                  (short)0, c, false, false);
    }

    // D layout: VGPR r -> row (r + 8*half), col l15
    const int mbase = row0 + half * 8;
    float* out = H + (size_t)mbase * FEAT + wave * 16 + l15;
    #pragma unroll
    for (int r = 0; r < 8; ++r) {
        if (mbase + r < n) out[(size_t)r * FEAT] = c[r];
    }
}

// ---------------- aggregation ----------------
// agg[i] = dis[i]^2 * h[i]   (self-loop term, also zero-initializes accumulator)
__global__ void agg_init(const float* __restrict__ H, const float* __restrict__ dis,
                         float* __restrict__ agg, int n) {
    int idx = blockIdx.x * blockDim.x + threadIdx.x;   // n*32 float4 elements
    if (idx >= n * (FEAT / 4)) return;
    int node = idx >> 5;
    float s = dis[node]; s = s * s;
    float4 v = ((const float4*)H)[idx];
    v.x *= s; v.y *= s; v.z *= s; v.w *= s;
    ((float4*)agg)[idx] = v;
}

// one wave32 per edge: float4 gather of h[src], scaled scatter-add into agg[dst]
__global__ void __launch_bounds__(256) agg_edges(const float* __restrict__ H,
                                                 const float* __restrict__ dis,
                                                 const long long* __restrict__ src,
                                                 const long long* __restrict__ dst,
                                                 float* __restrict__ agg, int e) {
    int widx = (blockIdx.x * blockDim.x + threadIdx.x) >> 5;
    int lane = threadIdx.x & 31;
    if (widx >= e) return;
    int s = (int)src[widx];
    int d = (int)dst[widx];
    float norm = dis[s] * dis[d];
    float4 v = ((const float4*)(H + (size_t)s * FEAT))[lane];
    float* out = agg + (size_t)d * FEAT + lane * 4;
    atomicAdd(out + 0, v.x * norm);
    atomicAdd(out + 1, v.y * norm);
    atomicAdd(out + 2, v.z * norm);
    atomicAdd(out + 3, v.w * norm);
}

// ---------------- batchnorm + prelu ----------------
__global__ void zero_stats(float* __restrict__ stats) {
    stats[threadIdx.x] = 0.0f;   // 256 threads: sum[128] + sumsq[128]
}

__global__ void bn_stats(const float* __restrict__ Y, const float* __restrict__ bias,
                         float* __restrict__ stats, int n) {
    int f = threadIdx.x;          // 128 threads = features
    float b = bias[f];
    float s = 0.0f, ss = 0.0f;
    for (int r = blockIdx.x; r < n; r += gridDim.x) {
        float v = Y[(size_t)r * FEAT + f] + b;
        s += v; ss += v * v;
    }
    atomicAdd(&stats[f], s);
    atomicAdd(&stats[FEAT + f], ss);
}

__global__ void bn_apply_prelu(const float* __restrict__ Y, const float* __restrict__ bias,
                               const float* __restrict__ stats, const float* __restrict__ g,
                               const float* __restrict__ be, const float* __restrict__ aprm,
                               float* __restrict__ out, int n, float invn) {
    int idx = blockIdx.x * blockDim.x + threadIdx.x;
    if (idx >= n * FEAT) return;
    int f = idx & (FEAT - 1);
    float mean = stats[f] * invn;
    float var  = stats[FEAT + f] * invn - mean * mean;   // biased variance
    float v = (Y[idx] + bias[f] - mean) * rsqrtf(var + 1e-5f) * g[f] + be[f];
    float a = aprm[0];
    out[idx] = v > 0.0f ? v : a * v;
}

// ---------------- launch ----------------
extern "C" void kernel_launch(void* const* d_in, const int* in_sizes, int n_in,
                              void* d_out, int out_size, void* d_ws, size_t ws_size,
                              hipStream_t stream) {
    const int n = in_sizes[0] / FEAT;          // 100000
    const int e = in_sizes[1] / 2;             // 800000

    const float*     x   = (const float*)d_in[0];
    const long long* ei  = (const long long*)d_in[1];   // int64 [2,E]
    const long long* src = ei;
    const long long* dst = ei + e;
    const float* W1 = (const float*)d_in[2];
    const float* b1 = (const float*)d_in[3];
    const float* g1 = (const float*)d_in[4];
    const float* be1 = (const float*)d_in[5];
    const float* W2 = (const float*)d_in[6];
    const float* b2 = (const float*)d_in[7];
    const float* g2 = (const float*)d_in[8];
    const float* be2 = (const float*)d_in[9];
    const float* a  = (const float*)d_in[10];

    float* buf0  = (float*)d_ws;               // [n,128]
    float* buf1  = buf0 + (size_t)n * FEAT;    // [n,128]
    float* dis   = buf1 + (size_t)n * FEAT;    // [n]
    float* stats = dis + n;                    // [256]
    float* yout  = (float*)d_out;

    const float invn = 1.0f / (float)n;
    const int TB = 256;
    const int gN    = (n + TB - 1) / TB;
    const int gE    = (e + TB - 1) / TB;
    const int gGemm = (n + 15) / 16;
    const int gAggI = (n * (FEAT / 4) + TB - 1) / TB;
    const int gAggE = ((size_t)e * 32 + TB - 1) / TB;
    const int gFull = (n * FEAT + TB - 1) / TB;

    // symmetric normalization: deg -> deg^{-1/2}
    deg_init<<<gN, TB, 0, stream>>>(dis, n);
    deg_accum<<<gE, TB, 0, stream>>>(dst, dis, e);
    deg_finalize<<<gN, TB, 0, stream>>>(dis, n);

    // ---- layer 1 ----
    gemm_wmma_f32<<<gGemm, TB, 0, stream>>>(x, W1, buf0, n);
    agg_init<<<gAggI, TB, 0, stream>>>(buf0, dis, buf1, n);
    agg_edges<<<gAggE, TB, 0, stream>>>(buf0, dis, src, dst, buf1, e);
    zero_stats<<<1, 256, 0, stream>>>(stats);
    bn_stats<<<1024, FEAT, 0, stream>>>(buf1, b1, stats, n);
    bn_apply_prelu<<<gFull, TB, 0, stream>>>(buf1, b1, stats, g1, be1, a, buf0, n, invn);

    // ---- layer 2 ----
    gemm_wmma_f32<<<gGemm, TB, 0, stream>>>(buf0, W2, buf1, n);
    agg_init<<<gAggI, TB, 0, stream>>>(buf1, dis, buf0, n);
    agg_edges<<<gAggE, TB, 0, stream>>>(buf1, dis, src, dst, buf0, e);
    zero_stats<<<1, 256, 0, stream>>>(stats);
    bn_stats<<<1024, FEAT, 0, stream>>>(buf0, b2, stats, n);
    bn_apply_prelu<<<gFull, TB, 0, stream>>>(buf0, b2, stats, g2, be2, a, yout, n, invn);
}